// GatNet_64991445123446
// MI455X (gfx1250) — compile-verified
//
#include <hip/hip_runtime.h>
#include <hip/hip_bf16.h>
#include <math.h>

typedef __attribute__((ext_vector_type(2))) float v2f;
typedef __attribute__((ext_vector_type(8))) float v8f;

#define HID   128
#define HEADS 8
#define CPH   16
#define NEG_SLOPE 0.2f

// ---------------- float atomic max (int-trick; m initialized to -inf) --------
__device__ __forceinline__ void atomicMaxF32(float* addr, float val) {
  if (val >= 0.0f) atomicMax((int*)addr, __float_as_int(val));
  else             atomicMin((unsigned int*)addr, (unsigned int)__float_as_int(val));
}

// ---------------- workspace init --------------------------------------------
__global__ void fill_f32(float* __restrict__ p, float v, int n) {
  int i = blockIdx.x * blockDim.x + threadIdx.x;
  if (i < n) p[i] = v;
}

// ---------------- C[M x 128] = A[M x K] @ B[K x 128], fp32 WMMA --------------
// grid.x = M/16 blocks, 256 threads = 8 waves; wave w owns N-tile w (16 cols).
// V_WMMA_F32_16X16X4_F32: A 16x4 (2 VGPR), B 4x16 (2 VGPR), C/D 16x16 (8 VGPR).
__global__ void gemm_wmma_f32(const float* __restrict__ A,
                              const float* __restrict__ B,
                              float* __restrict__ C, int K) {
  const int lane = threadIdx.x & 31;
  const int wave = threadIdx.x >> 5;        // 0..7 -> N tile index
  const int m0   = blockIdx.x * 16;
  const int n0   = wave * 16;
  const int r    = lane & 15;               // row (A) / col (B,C) within tile
  const int kh   = (lane >> 4) << 1;        // lanes 0-15: K {0,1}; 16-31: {2,3}

  const float* arow = A + (size_t)(m0 + r) * K;
  v8f acc = {0.f, 0.f, 0.f, 0.f, 0.f, 0.f, 0.f, 0.f};

  for (int k = 0; k < K; k += 4) {
    v2f a = *(const v2f*)(arow + k + kh);               // contiguous, 8B aligned
    v2f b;
    b.x = B[(size_t)(k + kh    ) * HID + n0 + r];
    b.y = B[(size_t)(k + kh + 1) * HID + n0 + r];
    acc = __builtin_amdgcn_wmma_f32_16x16x4_f32(false, a, false, b,
                                                (short)0, acc, false, false);
  }

  // C/D layout: VGPR j -> row m0+j (lanes 0-15) / m0+8+j (lanes 16-31), col n0+r
  const int crow = m0 + ((lane >> 4) << 3);
  float* cp = C + (size_t)crow * HID + n0 + r;
  #pragma unroll
  for (int j = 0; j < 8; ++j) cp[(size_t)j * HID] = acc[j];
}

// ---------------- al_src/al_dst: [N,H] einsum('nhc,hc->nh') ------------------
__global__ void attn_logits(const float* __restrict__ h,
                            const float* __restrict__ a_src,
                            const float* __restrict__ a_dst,
                            float* __restrict__ als, float* __restrict__ ald,
                            int N) {
  int idx = blockIdx.x * blockDim.x + threadIdx.x;   // n*HEADS + head
  if (idx >= N * HEADS) return;
  int n  = idx >> 3;
  int hd = idx & 7;
  const float* hp = h + (size_t)n * HID + hd * CPH;
  const float* as = a_src + hd * CPH;
  const float* ad = a_dst + hd * CPH;
  float s1 = 0.f, s2 = 0.f;
  #pragma unroll
  for (int c = 0; c < CPH; ++c) { float v = hp[c]; s1 += v * as[c]; s2 += v * ad[c]; }
  als[idx] = s1;
  ald[idx] = s2;
}

// ---------------- edge pass A: leaky-relu logits + segment max ---------------
__global__ void edge_logits_max(const long long* __restrict__ ei,
                                const float* __restrict__ als,
                                const float* __restrict__ ald,
                                float* __restrict__ ebuf, float* __restrict__ m,
                                int E, int N) {
  int e = blockIdx.x * blockDim.x + threadIdx.x;
  int Etot = E + N;
  if (e >= Etot) return;
  long long s, d;
  if (e < E) { s = ei[e]; d = ei[(size_t)E + e]; } else { s = d = e - E; }
  #pragma unroll
  for (int hd = 0; hd < HEADS; ++hd) {
    float v = als[(size_t)s * HEADS + hd] + ald[(size_t)d * HEADS + hd];
    v = (v > 0.f) ? v : NEG_SLOPE * v;
    ebuf[(size_t)e * HEADS + hd] = v;
    atomicMaxF32(&m[(size_t)d * HEADS + hd], v);
  }
}

// ---------------- edge pass B: p = exp(e - m[dst]) + segment sum -------------
__global__ void edge_expsum(const long long* __restrict__ ei,
                            const float* __restrict__ m,
                            float* __restrict__ ebuf, float* __restrict__ ssum,
                            int E, int N) {
  int e = blockIdx.x * blockDim.x + threadIdx.x;
  int Etot = E + N;
  if (e >= Etot) return;
  long long d;
  if (e < E) { d = ei[(size_t)E + e]; } else { d = e - E; }
  #pragma unroll
  for (int hd = 0; hd < HEADS; ++hd) {
    float p = __expf(ebuf[(size_t)e * HEADS + hd] - m[(size_t)d * HEADS + hd]);
    ebuf[(size_t)e * HEADS + hd] = p;
    atomicAdd(&ssum[(size_t)d * HEADS + hd], p);
  }
}

// ---------------- edge pass C: agg[dst] += h[src] * alpha --------------------
// one thread per (edge, head): 16 atomic f32 adds each
__global__ void edge_aggregate(const long long* __restrict__ ei,
                               const float* __restrict__ hfeat,
                               const float* __restrict__ ebuf,
                               const float* __restrict__ ssum,
                               float* __restrict__ agg, int E, int N) {
  int idx = blockIdx.x * blockDim.x + threadIdx.x;   // e*HEADS + head
  int Etot = E + N;
  if (idx >= Etot * HEADS) return;
  int e  = idx >> 3;
  int hd = idx & 7;
  long long s, d;
  if (e < E) { s = ei[e]; d = ei[(size_t)E + e]; } else { s = d = e - E; }
  float alpha = ebuf[(size_t)e * HEADS + hd] / ssum[(size_t)d * HEADS + hd];
  const float* hs = hfeat + (size_t)s * HID + hd * CPH;
  float* od = agg + (size_t)d * HID + hd * CPH;
  #pragma unroll
  for (int c = 0; c < CPH; ++c) atomicAdd(&od[c], hs[c] * alpha);
}

// ---------------- finalize: out = elu(agg + b) -------------------------------
__global__ void finalize_elu(const float* __restrict__ agg,
                             const float* __restrict__ b,
                             float* __restrict__ out, int N) {
  int i = blockIdx.x * blockDim.x + threadIdx.x;
  if (i >= N * HID) return;
  float v = agg[i] + b[i & (HID - 1)];
  out[i] = (v > 0.f) ? v : expm1f(v);
}

// ---------------- final fc: out[n] = h[n] . fc_w + fc_b ----------------------
__global__ void fc_out(const float* __restrict__ h, const float* __restrict__ w,
                       const float* __restrict__ bias, float* __restrict__ out,
                       int N) {
  int n = blockIdx.x * blockDim.x + threadIdx.x;
  if (n >= N) return;
  const float* hp = h + (size_t)n * HID;
  float acc = 0.f;
  #pragma unroll 8
  for (int f = 0; f < HID; ++f) acc += hp[f] * w[f];
  out[n] = acc + bias[0];
}

extern "C" void kernel_launch(void* const* d_in, const int* in_sizes, int n_in,
                              void* d_out, int out_size, void* d_ws, size_t ws_size,
                              hipStream_t stream) {
  const float*     x   = (const float*)d_in[0];
  const long long* ei  = (const long long*)d_in[1];
  const float* W[3]  = {(const float*)d_in[2], (const float*)d_in[6],  (const float*)d_in[10]};
  const float* aS[3] = {(const float*)d_in[3], (const float*)d_in[7],  (const float*)d_in[11]};
  const float* aD[3] = {(const float*)d_in[4], (const float*)d_in[8],  (const float*)d_in[12]};
  const float* bb[3] = {(const float*)d_in[5], (const float*)d_in[9],  (const float*)d_in[13]};
  const float* fcw = (const float*)d_in[14];
  const float* fcb = (const float*)d_in[15];

  const int N    = in_sizes[0] / 16;   // 50000 (multiple of 16)
  const int E    = in_sizes[1] / 2;    // 800000
  const int Etot = E + N;

  // workspace carve-up (all float32)
  float* ws    = (float*)d_ws;
  float* hFeat = ws;  ws += (size_t)N * HID;     // GEMM output h
  float* agg   = ws;  ws += (size_t)N * HID;     // attention-weighted aggregate
  float* hIn   = ws;  ws += (size_t)N * HID;     // elu(agg+b) -> next layer input
  float* als   = ws;  ws += (size_t)N * HEADS;
  float* ald   = ws;  ws += (size_t)N * HEADS;
  float* mmax  = ws;  ws += (size_t)N * HEADS;
  float* ssum  = ws;  ws += (size_t)N * HEADS;
  float* ebuf  = ws;  ws += (size_t)Etot * HEADS;

  const int TB = 256;
  const float* in = x;
  int K = 16;

  for (int l = 0; l < 3; ++l) {
    // re-init per layer (deterministic across graph replays)
    fill_f32<<<(N * HEADS + TB - 1) / TB, TB, 0, stream>>>(mmax, -INFINITY, N * HEADS);
    fill_f32<<<(N * HEADS + TB - 1) / TB, TB, 0, stream>>>(ssum, 0.f,       N * HEADS);
    fill_f32<<<(N * HID   + TB - 1) / TB, TB, 0, stream>>>(agg,  0.f,       N * HID);

    gemm_wmma_f32<<<N / 16, TB, 0, stream>>>(in, W[l], hFeat, K);
    attn_logits<<<(N * HEADS + TB - 1) / TB, TB, 0, stream>>>(hFeat, aS[l], aD[l],
                                                              als, ald, N);
    edge_logits_max<<<(Etot + TB - 1) / TB, TB, 0, stream>>>(ei, als, ald, ebuf,
                                                             mmax, E, N);
    edge_expsum<<<(Etot + TB - 1) / TB, TB, 0, stream>>>(ei, mmax, ebuf, ssum, E, N);
    edge_aggregate<<<((size_t)Etot * HEADS + TB - 1) / TB, TB, 0, stream>>>(
        ei, hFeat, ebuf, ssum, agg, E, N);
    finalize_elu<<<(N * HID + TB - 1) / TB, TB, 0, stream>>>(agg, bb[l], hIn, N);

    in = hIn;
    K  = HID;
  }

  fc_out<<<(N + TB - 1) / TB, TB, 0, stream>>>(hIn, fcw, fcb, (float*)d_out, N);
}